// AdaFace_79663053406375
// MI455X (gfx1250) — compile-verified
//
#include <hip/hip_runtime.h>
#include <hip/hip_bf16.h>

// AdaFace margin kernel for MI455X (gfx1250, wave32).
// out[b,c] = (cos(acos(cos_bc) + onehot*g_ang_b) - onehot*g_add_b) * S
// Non-label columns reduce to cos_bc * S  ->  pure HBM-streaming multiply
// (~351 MB traffic, ~15 us floor at 23.3 TB/s). Only the <=512 label
// elements need the transcendental path.

#define B_ROWS 512
#define C_COLS 85742
#define M_CONST 0.4f
#define H_CONST 0.333f
#define S_CONST 64.0f
#define T_ALPHA 0.01f
#define EPS_CONST 0.001f

typedef float v4f __attribute__((ext_vector_type(4)));

// ---------------------------------------------------------------------------
// Kernel 1: norm statistics + per-row margin params (one 512-thread block).
// rowp[2b] = g_angular(b), rowp[2b+1] = g_add(b)
// ---------------------------------------------------------------------------
__global__ __launch_bounds__(B_ROWS) void adaface_stats_kernel(
    const float* __restrict__ norms,
    const float* __restrict__ batch_mean,
    const float* __restrict__ batch_std,
    float* __restrict__ rowp) {
  __shared__ float s_sum[B_ROWS];
  __shared__ float s_sq[B_ROWS];

  const int t = threadIdx.x;
  float x = norms[t];
  x = fminf(fmaxf(x, 0.001f), 100.0f);  // safe_norms (clip)
  s_sum[t] = x;
  s_sq[t] = x * x;
  __syncthreads();

  // Pairwise tree reduction (sum and sum of squares).
  for (int off = B_ROWS / 2; off > 0; off >>= 1) {
    if (t < off) {
      s_sum[t] += s_sum[t + off];
      s_sq[t] += s_sq[t + off];
    }
    __syncthreads();
  }

  const float sum = s_sum[0];
  const float sumsq = s_sq[0];
  const float mean = sum * (1.0f / (float)B_ROWS);
  // Unbiased (ddof=1) variance: (sumsq - sum*mean) / (n-1)
  float var = (sumsq - sum * mean) * (1.0f / (float)(B_ROWS - 1));
  var = fmaxf(var, 0.0f);
  const float stdv = sqrtf(var);

  const float new_mean = mean * T_ALPHA + (1.0f - T_ALPHA) * batch_mean[0];
  const float new_std = stdv * T_ALPHA + (1.0f - T_ALPHA) * batch_std[0];

  float ms = (x - new_mean) / (new_std + EPS_CONST) * H_CONST;
  ms = fminf(fmaxf(ms, -1.0f), 1.0f);

  rowp[2 * t] = -M_CONST * ms;               // g_angular
  rowp[2 * t + 1] = M_CONST + M_CONST * ms;  // g_add
}

// ---------------------------------------------------------------------------
// Kernel 2: the bandwidth kernel. out = cosine * S, flat float4 stream.
// Read-once / write-once, 351 MB > 192 MB L2 -> non-temporal hints
// (global_load_b128 / global_store_b128 with TH=NT). Each thread owns 4
// grid-strided float4s: 4 independent NT loads issued back-to-back (4 b128
// in flight per thread), one wait, 4 NT stores.
// ---------------------------------------------------------------------------
__global__ __launch_bounds__(256) void adaface_scale_kernel(
    const v4f* __restrict__ in, v4f* __restrict__ out,
    unsigned int n4, unsigned int total_threads) {
  const unsigned int tid = blockIdx.x * 256u + threadIdx.x;
  const unsigned int i0 = tid;
  const unsigned int i1 = tid + total_threads;
  const unsigned int i2 = tid + 2u * total_threads;
  const unsigned int i3 = tid + 3u * total_threads;

  if (i3 < n4) {
    // Fast path: 4 unguarded NT b128 loads, then 4 NT b128 stores.
    v4f a = __builtin_nontemporal_load(in + i0);
    v4f b = __builtin_nontemporal_load(in + i1);
    v4f c = __builtin_nontemporal_load(in + i2);
    v4f d = __builtin_nontemporal_load(in + i3);
    a *= S_CONST;
    b *= S_CONST;
    c *= S_CONST;
    d *= S_CONST;
    __builtin_nontemporal_store(a, out + i0);
    __builtin_nontemporal_store(b, out + i1);
    __builtin_nontemporal_store(c, out + i2);
    __builtin_nontemporal_store(d, out + i3);
  } else {
    // Tail path (last ~256 threads of the grid only).
    for (unsigned int i = i0; i < n4; i += total_threads) {
      v4f v = __builtin_nontemporal_load(in + i);
      v *= S_CONST;
      __builtin_nontemporal_store(v, out + i);
    }
  }
}

// ---------------------------------------------------------------------------
// Kernel 3: overwrite the single label element per valid row with the exact
// transcendental margin path. Runs after kernel 2 (stream order).
// ---------------------------------------------------------------------------
__global__ __launch_bounds__(256) void adaface_fixup_kernel(
    const float* __restrict__ cosine,
    const int* __restrict__ label,
    const float* __restrict__ rowp,
    float* __restrict__ out) {
  const int b = blockIdx.x * blockDim.x + threadIdx.x;
  if (b >= B_ROWS) return;
  const int lb = label[b];
  if (lb < 0 || lb >= C_COLS) return;  // label == -1 -> no margin applied
  const size_t idx = (size_t)b * (size_t)C_COLS + (size_t)lb;
  const float c = cosine[idx];
  const float g_ang = rowp[2 * b];
  const float g_add = rowp[2 * b + 1];
  const float theta = acosf(c) + g_ang;
  out[idx] = (cosf(theta) - g_add) * S_CONST;
}

extern "C" void kernel_launch(void* const* d_in, const int* in_sizes, int n_in,
                              void* d_out, int out_size, void* d_ws, size_t ws_size,
                              hipStream_t stream) {
  (void)in_sizes; (void)n_in; (void)out_size; (void)ws_size;

  const float* d_cosine = (const float*)d_in[0];  // [B, C] fp32
  const float* d_norms = (const float*)d_in[1];   // [B, 1] fp32
  const float* d_bmean = (const float*)d_in[2];   // [1] fp32
  const float* d_bstd = (const float*)d_in[3];    // [1] fp32
  const int* d_label = (const int*)d_in[4];       // [B] int32
  float* d_o = (float*)d_out;                     // [B, C] fp32
  float* d_rowp = (float*)d_ws;                   // [2*B] fp32 scratch

  // 1) Per-row margin parameters (tiny).
  adaface_stats_kernel<<<1, B_ROWS, 0, stream>>>(d_norms, d_bmean, d_bstd, d_rowp);

  // 2) Streaming multiply: 43,899,904 fp32 = 10,974,976 float4 (exact /4).
  const unsigned int n4 = (unsigned int)(((size_t)B_ROWS * (size_t)C_COLS) / 4);
  const unsigned int threads = 256;
  const unsigned int per_thread = 4;  // 4 NT b128 loads in flight per thread
  const unsigned int blocks =
      (n4 + threads * per_thread - 1) / (threads * per_thread);
  const unsigned int total_threads = blocks * threads;
  adaface_scale_kernel<<<blocks, threads, 0, stream>>>(
      (const v4f*)d_cosine, (v4f*)d_o, n4, total_threads);

  // 3) Overwrite the label elements with the exact margin path.
  adaface_fixup_kernel<<<(B_ROWS + 255) / 256, 256, 0, stream>>>(
      d_cosine, d_label, d_rowp, d_o);
}